// SparseNet_83399674954375
// MI455X (gfx1250) — compile-verified
//
#include <hip/hip_runtime.h>
#include <math.h>

// ---------------- problem constants ----------------
#define NN      50000
#define EE      800000
#define INC     64
#define HID     128
#define K1C     64
#define K2C     16
#define THRV    (1.0f / 63.0f)

// output layout (floats)
#define S1_OFF   4
#define A1_OFF   (4 + NN * K1C)            // 3200004
#define S2_OFF   (A1_OFF + K1C * K1C)      // 3204100
#define ADJ2_OFF (S2_OFF + K1C * K2C)      // 3205124

// workspace layout (float offsets)
#define WS_AGG1    ((size_t)0)                      // NN*64
#define WS_AS1     ((size_t)3200000)                // NN*64
#define WS_DEG1    ((size_t)6400000)                // NN
#define WS_POOLX   ((size_t)6450000)                // 64*128
#define WS_POOLADJ ((size_t)6458192)                // 64*64
#define WS_STS     ((size_t)6462288)                // 64*64
#define WS_DEN1    ((size_t)6466384)                // 16 (pad)
#define WS_ZERO_END ((size_t)6466400)               // zeroed region length
#define WS_X1      ((size_t)6466400)                // NN*128
#define WS_S1SOFT  ((size_t)12866400)               // NN*64
#define WS_S1LIN   ((size_t)16066400)               // NN*64
#define WS_A1F     ((size_t)19266400)               // 64*64
#define WS_X2      ((size_t)19270496)               // 64*128
#define WS_S2SOFT  ((size_t)19278688)               // 64*16

typedef float v2f __attribute__((ext_vector_type(2)));
typedef float v8f __attribute__((ext_vector_type(8)));

__device__ __forceinline__ v8f wmma4(v2f a, v2f b, v8f c) {
    return __builtin_amdgcn_wmma_f32_16x16x4_f32(
        /*neg_a=*/false, a, /*neg_b=*/false, b,
        /*c_mod=*/(short)0, c, /*reuse_a=*/false, /*reuse_b=*/false);
}

__device__ __forceinline__ float wave_sum32(float v) {
    #pragma unroll
    for (int o = 16; o > 0; o >>= 1) v += __shfl_xor(v, o, 32);
    return v;
}
__device__ __forceinline__ float wave_max32(float v) {
    #pragma unroll
    for (int o = 16; o > 0; o >>= 1) v = fmaxf(v, __shfl_xor(v, o, 32));
    return v;
}
__device__ __forceinline__ float block_reduce_sum(float v, float* sh) {
    int tid = threadIdx.x;
    sh[tid] = v; __syncthreads();
    for (int s = 128; s > 0; s >>= 1) {
        if (tid < s) sh[tid] += sh[tid + s];
        __syncthreads();
    }
    float r = sh[0]; __syncthreads();
    return r;
}

// ---------------- edge scatters ----------------
// out[sidx[e]] += feat[gidx[e]]   (64 features per edge)
__global__ __launch_bounds__(256) void mincut_scatter64_kernel(
    const int* __restrict__ gidx, const int* __restrict__ sidx,
    const float* __restrict__ feat, float* __restrict__ out) {
    long long i = (long long)blockIdx.x * 256 + threadIdx.x;
    long long e = i >> 6;
    int c = (int)(i & 63);
    if (e < EE) {
        int g = gidx[e], s = sidx[e];
        atomicAdd(&out[(size_t)s * 64 + c], feat[(size_t)g * 64 + c]);
    }
}

__global__ __launch_bounds__(256) void mincut_deg_kernel(
    const int* __restrict__ src, float* __restrict__ deg) {
    int e = blockIdx.x * 256 + threadIdx.x;
    if (e < EE) atomicAdd(&deg[src[e]], 1.0f);
}

// ---------------- x1 = relu(agg1@W1_rel + x@W1_root + b1) ----------------
// one wave per 16-row strip, WMMA f32 16x16x4, K=64, 8 col tiles of 16
__global__ __launch_bounds__(256) void mincut_gemm_x1_kernel(
    const float* __restrict__ agg1, const float* __restrict__ x,
    const float* __restrict__ W1rel, const float* __restrict__ W1root,
    const float* __restrict__ b1, float* __restrict__ x1) {
    int wave = (blockIdx.x * 256 + threadIdx.x) >> 5;
    int lane = threadIdx.x & 31;
    int hi = lane >> 4, lr = lane & 15;
    if (wave >= NN / 16) return;               // wave-uniform exit
    int m0 = wave * 16;

    v2f aA[16], aX[16];
    #pragma unroll
    for (int kk = 0; kk < 16; ++kk) {
        int k0 = kk * 4 + hi * 2;
        const float* pa = agg1 + (size_t)(m0 + lr) * 64 + k0;
        aA[kk] = (v2f){pa[0], pa[1]};
        const float* px = x + (size_t)(m0 + lr) * 64 + k0;
        aX[kk] = (v2f){px[0], px[1]};
    }
    for (int t = 0; t < 8; ++t) {
        int n0 = t * 16;
        v8f acc = (v8f){0.f, 0.f, 0.f, 0.f, 0.f, 0.f, 0.f, 0.f};
        #pragma unroll
        for (int kk = 0; kk < 16; ++kk) {
            int kr = kk * 4 + hi * 2;
            v2f b = (v2f){W1rel[(size_t)kr * HID + n0 + lr],
                          W1rel[(size_t)(kr + 1) * HID + n0 + lr]};
            acc = wmma4(aA[kk], b, acc);
        }
        #pragma unroll
        for (int kk = 0; kk < 16; ++kk) {
            int kr = kk * 4 + hi * 2;
            v2f b = (v2f){W1root[(size_t)kr * HID + n0 + lr],
                          W1root[(size_t)(kr + 1) * HID + n0 + lr]};
            acc = wmma4(aX[kk], b, acc);
        }
        float bias = b1[n0 + lr];
        #pragma unroll
        for (int j = 0; j < 8; ++j) {
            int r = m0 + j + 8 * hi;
            float v = acc[j] + bias;
            x1[(size_t)r * HID + n0 + lr] = v > 0.f ? v : 0.f;
        }
    }
}

// ---------------- s1_lin = x1 @ Wp1  (K=128, 4 col tiles) ----------------
__global__ __launch_bounds__(256) void mincut_gemm_s1_kernel(
    const float* __restrict__ x1, const float* __restrict__ Wp1,
    float* __restrict__ s1lin) {
    int wave = (blockIdx.x * 256 + threadIdx.x) >> 5;
    int lane = threadIdx.x & 31;
    int hi = lane >> 4, lr = lane & 15;
    if (wave >= NN / 16) return;
    int m0 = wave * 16;

    v2f aA[32];
    #pragma unroll
    for (int kk = 0; kk < 32; ++kk) {
        int k0 = kk * 4 + hi * 2;
        const float* pa = x1 + (size_t)(m0 + lr) * HID + k0;
        aA[kk] = (v2f){pa[0], pa[1]};
    }
    for (int t = 0; t < 4; ++t) {
        int n0 = t * 16;
        v8f acc = (v8f){0.f, 0.f, 0.f, 0.f, 0.f, 0.f, 0.f, 0.f};
        #pragma unroll
        for (int kk = 0; kk < 32; ++kk) {
            int kr = kk * 4 + hi * 2;
            v2f b = (v2f){Wp1[(size_t)kr * K1C + n0 + lr],
                          Wp1[(size_t)(kr + 1) * K1C + n0 + lr]};
            acc = wmma4(aA[kk], b, acc);
        }
        #pragma unroll
        for (int j = 0; j < 8; ++j) {
            int r = m0 + j + 8 * hi;
            s1lin[(size_t)r * K1C + n0 + lr] = acc[j];
        }
    }
}

// ---------------- LN + softmax over 64 cols, one wave per row ----------------
__global__ __launch_bounds__(256) void mincut_ln_softmax1_kernel(
    const float* __restrict__ s1lin, const float* __restrict__ bp1,
    const float* __restrict__ lnw, const float* __restrict__ lnb,
    float* __restrict__ s1_out, float* __restrict__ s1soft) {
    int row = (blockIdx.x * 256 + threadIdx.x) >> 5;
    int lane = threadIdx.x & 31;
    if (row >= NN) return;
    const float* p = s1lin + (size_t)row * 64;
    float v0 = p[lane] + bp1[lane];
    float v1 = p[lane + 32] + bp1[lane + 32];
    float m = wave_sum32(v0 + v1) * (1.0f / 64.0f);
    float d0 = v0 - m, d1 = v1 - m;
    float var = wave_sum32(d0 * d0 + d1 * d1) * (1.0f / 64.0f);
    float inv = 1.0f / sqrtf(var + 1e-5f);
    float y0 = d0 * inv * lnw[lane] + lnb[lane];
    float y1 = d1 * inv * lnw[lane + 32] + lnb[lane + 32];
    s1_out[(size_t)row * 64 + lane] = y0;
    s1_out[(size_t)row * 64 + lane + 32] = y1;
    float mx = wave_max32(fmaxf(y0, y1));
    float e0 = expf(y0 - mx), e1 = expf(y1 - mx);
    float r = 1.0f / wave_sum32(e0 + e1);
    s1soft[(size_t)row * 64 + lane] = e0 * r;
    s1soft[(size_t)row * 64 + lane + 32] = e1 * r;
}

// ---------------- pooled reductions: sT@x1, sT@A_s1, sT@s (split-K WMMA) ----
__global__ __launch_bounds__(256) void mincut_pooled_kernel(
    const float* __restrict__ s, const float* __restrict__ x1,
    const float* __restrict__ As1, float* __restrict__ poolx,
    float* __restrict__ pooladj, float* __restrict__ sts) {
    int w = (blockIdx.x * 256 + threadIdx.x) >> 5;   // 1600 waves
    int lane = threadIdx.x & 31;
    int hi = lane >> 4, lr = lane & 15;
    int job = w & 63;
    int slice = w >> 6;                              // 25 slices * 2000 nodes
    const float* B; float* C; int ldB, ldC, m0, n0;
    if (job < 32)      { B = x1;  C = poolx;   ldB = HID; ldC = HID; m0 = (job >> 3) * 16; n0 = (job & 7) * 16; }
    else if (job < 48) { int j = job - 32; B = As1; C = pooladj; ldB = 64; ldC = 64; m0 = (j >> 2) * 16; n0 = (j & 3) * 16; }
    else               { int j = job - 48; B = s;   C = sts;     ldB = 64; ldC = 64; m0 = (j >> 2) * 16; n0 = (j & 3) * 16; }
    int nb = slice * 2000, ne = nb + 2000;
    v8f acc = (v8f){0.f, 0.f, 0.f, 0.f, 0.f, 0.f, 0.f, 0.f};
    for (int node = nb; node < ne; node += 4) {
        int kr = node + hi * 2;
        v2f a = (v2f){s[(size_t)kr * 64 + m0 + lr],
                      s[(size_t)(kr + 1) * 64 + m0 + lr]};
        v2f b = (v2f){B[(size_t)kr * ldB + n0 + lr],
                      B[(size_t)(kr + 1) * ldB + n0 + lr]};
        acc = wmma4(a, b, acc);
    }
    #pragma unroll
    for (int j = 0; j < 8; ++j)
        atomicAdd(&C[(size_t)(m0 + j + 8 * hi) * ldC + n0 + lr], acc[j]);
}

// ---------------- den1 = sum_i deg[i] * ||s_i||^2 ----------------
__global__ __launch_bounds__(256) void mincut_den1_kernel(
    const float* __restrict__ s, const float* __restrict__ deg,
    float* __restrict__ den) {
    __shared__ float sh[256];
    int i = blockIdx.x * 256 + threadIdx.x;
    float part = 0.f;
    if (i < NN) {
        const float* p = s + (size_t)i * 64;
        float a = 0.f;
        for (int k = 0; k < 64; ++k) a += p[k] * p[k];
        part = deg[i] * a;
    }
    float r = block_reduce_sum(part, sh);
    if (threadIdx.x == 0) atomicAdd(den, r);
}

// ---------------- finalize layer-1 scalars, A1 ----------------
__global__ __launch_bounds__(256) void mincut_finalize1_kernel(
    const float* __restrict__ pooladj, const float* __restrict__ sts,
    const float* __restrict__ den1, float* __restrict__ out,
    float* __restrict__ A1f) {
    __shared__ float sh[256];
    __shared__ float inv[64];
    int tid = threadIdx.x;
    float t = (tid < 64) ? pooladj[tid * 65] : 0.f;
    float tr = block_reduce_sum(t, sh);
    float np = 0.f;
    for (int i = tid; i < 4096; i += 256) { float v = sts[i]; np += v * v; }
    float nrm = sqrtf(block_reduce_sum(np, sh));
    float op = 0.f;
    for (int i = tid; i < 4096; i += 256) {
        int r = i >> 6, c = i & 63;
        float v = sts[i] / (nrm + 1e-10f) - ((r == c) ? 0.125f : 0.f);
        op += v * v;
    }
    float o1 = sqrtf(block_reduce_sum(op, sh));
    if (tid < 64) {
        float rs = 0.f;
        for (int j = 0; j < 64; ++j) if (j != tid) rs += pooladj[tid * 64 + j];
        inv[tid] = 1.0f / (sqrtf(rs) + 1e-15f);
    }
    __syncthreads();
    for (int i = tid; i < 4096; i += 256) {
        int r = i >> 6, c = i & 63;
        float v = (r == c) ? 0.f : pooladj[i];
        float an = v * inv[r] * inv[c];
        float a1 = (an > THRV) ? 1.0f : 0.0f;
        out[A1_OFF + i] = a1;
        A1f[i] = a1;
    }
    if (tid == 0) { out[0] = -tr / (den1[0] + 1e-10f); out[1] = o1; }
}

// ---------------- layer 2: x2 = relu(A1ᵀ@poolx @W2rel + poolx@W2root + b2) --
__global__ __launch_bounds__(256) void mincut_x2_kernel(
    const float* __restrict__ A1f, const float* __restrict__ poolx,
    const float* __restrict__ W2rel, const float* __restrict__ W2root,
    const float* __restrict__ b2, float* __restrict__ x2) {
    __shared__ float agg2[K1C * HID];
    int tid = threadIdx.x;
    for (int idx = tid; idx < K1C * HID; idx += 256) {
        int i = idx >> 7, c = idx & 127;
        float a = 0.f;
        for (int r = 0; r < 64; ++r) a += A1f[r * 64 + i] * poolx[r * HID + c];
        agg2[idx] = a;
    }
    __syncthreads();
    for (int idx = tid; idx < K1C * HID; idx += 256) {
        int i = idx >> 7, c = idx & 127;
        float v = b2[c];
        for (int k = 0; k < HID; ++k)
            v += agg2[i * HID + k] * W2rel[k * HID + c]
               + poolx[i * HID + k] * W2root[k * HID + c];
        x2[idx] = v > 0.f ? v : 0.f;
    }
}

// ---------------- s2 = LN(x2@Wp2 + bp2), softmax ----------------
__global__ __launch_bounds__(64) void mincut_s2_kernel(
    const float* __restrict__ x2, const float* __restrict__ Wp2,
    const float* __restrict__ bp2, const float* __restrict__ lnw,
    const float* __restrict__ lnb, float* __restrict__ s2_out,
    float* __restrict__ s2soft) {
    int row = threadIdx.x;
    if (row >= 64) return;
    float lin[16];
    for (int k = 0; k < 16; ++k) {
        float a = bp2[k];
        for (int j = 0; j < HID; ++j) a += x2[row * HID + j] * Wp2[j * 16 + k];
        lin[k] = a;
    }
    float m = 0.f;
    for (int k = 0; k < 16; ++k) m += lin[k];
    m *= (1.0f / 16.0f);
    float var = 0.f;
    for (int k = 0; k < 16; ++k) { float d = lin[k] - m; var += d * d; }
    var *= (1.0f / 16.0f);
    float inv = 1.0f / sqrtf(var + 1e-5f);
    float y[16], mx = -1e30f;
    for (int k = 0; k < 16; ++k) {
        y[k] = (lin[k] - m) * inv * lnw[k] + lnb[k];
        s2_out[row * 16 + k] = y[k];
        mx = fmaxf(mx, y[k]);
    }
    float ssum = 0.f;
    for (int k = 0; k < 16; ++k) { y[k] = expf(y[k] - mx); ssum += y[k]; }
    float r = 1.0f / ssum;
    for (int k = 0; k < 16; ++k) s2soft[row * 16 + k] = y[k] * r;
}

// ---------------- finalize layer-2 scalars, adj2 ----------------
__global__ __launch_bounds__(256) void mincut_finalize2_kernel(
    const float* __restrict__ A1f, const float* __restrict__ s2soft,
    float* __restrict__ out) {
    __shared__ float sh[256];
    __shared__ float As2[K1C * K2C];
    __shared__ float deg2[64];
    __shared__ float padj2[256];
    __shared__ float sts2[256];
    __shared__ float inv2[16];
    int tid = threadIdx.x;
    if (tid < 64) {
        float d = 0.f;
        for (int j = 0; j < 64; ++j) d += A1f[tid * 64 + j];
        deg2[tid] = d;
    }
    for (int idx = tid; idx < K1C * K2C; idx += 256) {
        int r = idx >> 4, k = idx & 15;
        float a = 0.f;
        for (int q = 0; q < 64; ++q) a += A1f[r * 64 + q] * s2soft[q * 16 + k];
        As2[idx] = a;
    }
    __syncthreads();
    { // padj2 = s2ᵀ @ As2 ; sts2 = s2ᵀ @ s2   (16x16 each, one elem/thread)
        int a = tid >> 4, b = tid & 15;
        float v = 0.f, wv = 0.f;
        for (int i = 0; i < 64; ++i) {
            float sa = s2soft[i * 16 + a];
            v += sa * As2[i * 16 + b];
            wv += sa * s2soft[i * 16 + b];
        }
        padj2[tid] = v; sts2[tid] = wv;
    }
    float dpart = 0.f;
    if (tid < 64) {
        float a = 0.f;
        for (int k = 0; k < 16; ++k) { float sv = s2soft[tid * 16 + k]; a += sv * sv; }
        dpart = deg2[tid] * a;
    }
    float den2 = block_reduce_sum(dpart, sh);
    float trp = (tid < 16) ? padj2[tid * 17] : 0.f;
    float tr2 = block_reduce_sum(trp, sh);
    float sv = sts2[tid];
    float nrm2 = sqrtf(block_reduce_sum(sv * sv, sh));
    int rr = tid >> 4, cc = tid & 15;
    float dv = sts2[tid] / (nrm2 + 1e-10f) - ((rr == cc) ? 0.25f : 0.f);
    float o2 = sqrtf(block_reduce_sum(dv * dv, sh));
    if (tid < 16) {
        float rs = 0.f;
        for (int j = 0; j < 16; ++j) if (j != tid) rs += padj2[tid * 16 + j];
        inv2[tid] = 1.0f / (sqrtf(rs) + 1e-15f);
    }
    __syncthreads();
    {
        float v = (rr == cc) ? 0.f : padj2[tid];
        out[ADJ2_OFF + tid] = v * inv2[rr] * inv2[cc];
    }
    if (tid == 0) { out[2] = -tr2 / (den2 + 1e-10f); out[3] = o2; }
}

// ---------------- launch ----------------
extern "C" void kernel_launch(void* const* d_in, const int* in_sizes, int n_in,
                              void* d_out, int out_size, void* d_ws, size_t ws_size,
                              hipStream_t stream) {
    const float* x      = (const float*)d_in[0];
    const int*   eidx   = (const int*)d_in[1];
    const float* W1rel  = (const float*)d_in[2];
    const float* b1rel  = (const float*)d_in[3];
    const float* W1root = (const float*)d_in[4];
    const float* Wp1    = (const float*)d_in[5];
    const float* bp1    = (const float*)d_in[6];
    const float* ln1w   = (const float*)d_in[7];
    const float* ln1b   = (const float*)d_in[8];
    const float* W2rel  = (const float*)d_in[9];
    const float* b2rel  = (const float*)d_in[10];
    const float* W2root = (const float*)d_in[11];
    const float* Wp2    = (const float*)d_in[12];
    const float* bp2    = (const float*)d_in[13];
    const float* ln2w   = (const float*)d_in[14];
    const float* ln2b   = (const float*)d_in[15];
    float* out = (float*)d_out;
    float* ws  = (float*)d_ws;

    const int* src = eidx;
    const int* dst = eidx + EE;

    float* agg1    = ws + WS_AGG1;
    float* As1     = ws + WS_AS1;
    float* deg1    = ws + WS_DEG1;
    float* poolx   = ws + WS_POOLX;
    float* pooladj = ws + WS_POOLADJ;
    float* sts     = ws + WS_STS;
    float* den1    = ws + WS_DEN1;
    float* x1      = ws + WS_X1;
    float* s1soft  = ws + WS_S1SOFT;
    float* s1lin   = ws + WS_S1LIN;
    float* A1f     = ws + WS_A1F;
    float* x2      = ws + WS_X2;
    float* s2soft  = ws + WS_S2SOFT;

    hipMemsetAsync(ws, 0, WS_ZERO_END * sizeof(float), stream);

    // agg1[dst] += x[src]; deg1[src] += 1
    mincut_scatter64_kernel<<<(EE * 64) / 256, 256, 0, stream>>>(src, dst, x, agg1);
    mincut_deg_kernel<<<EE / 256, 256, 0, stream>>>(src, deg1);

    // x1 = relu(agg1@W1rel + x@W1root + b1)   [WMMA]
    int nStripBlocks = (NN / 16 + 7) / 8;   // 8 waves per block
    mincut_gemm_x1_kernel<<<nStripBlocks, 256, 0, stream>>>(agg1, x, W1rel, W1root, b1rel, x1);

    // s1lin = x1 @ Wp1   [WMMA]
    mincut_gemm_s1_kernel<<<nStripBlocks, 256, 0, stream>>>(x1, Wp1, s1lin);

    // s1 (LN out -> d_out), s1soft
    mincut_ln_softmax1_kernel<<<NN / 8, 256, 0, stream>>>(s1lin, bp1, ln1w, ln1b,
                                                          out + S1_OFF, s1soft);

    // A_s1[src] += s1soft[dst]
    mincut_scatter64_kernel<<<(EE * 64) / 256, 256, 0, stream>>>(dst, src, s1soft, As1);

    // pooled_x, pooled_adj, sT_s   [WMMA split-K: 64 jobs x 25 slices = 1600 waves]
    mincut_pooled_kernel<<<200, 256, 0, stream>>>(s1soft, x1, As1, poolx, pooladj, sts);

    // den1
    mincut_den1_kernel<<<(NN + 255) / 256, 256, 0, stream>>>(s1soft, deg1, den1);

    // mc1, o1, A1
    mincut_finalize1_kernel<<<1, 256, 0, stream>>>(pooladj, sts, den1, out, A1f);

    // layer 2
    mincut_x2_kernel<<<1, 256, 0, stream>>>(A1f, poolx, W2rel, W2root, b2rel, x2);
    mincut_s2_kernel<<<1, 64, 0, stream>>>(x2, Wp2, bp2, ln2w, ln2b, out + S2_OFF, s2soft);
    mincut_finalize2_kernel<<<1, 256, 0, stream>>>(A1f, s2soft, out);
}